// DiffPoolBatchedGraphLayer_39986145526428
// MI455X (gfx1250) — compile-verified
//
#include <hip/hip_runtime.h>
#include <math.h>

// Problem constants (match reference setup_inputs)
#define N_NODES    20000
#define NPER       1000
#define NGRAPH     20
#define KPER       50
#define ASSIGNC    1000
#define ASSIGN_PAD 1024
#define DIN        128
#define DK         256     // 2*DIN
#define PCOLS      64      // padded compact p width (50 -> 64)

typedef float v2f __attribute__((ext_vector_type(2)));
typedef float v8f __attribute__((ext_vector_type(8)));

// D = A(16x4 f32) * B(4x16 f32) + C(16x16 f32), full-precision CDNA5 WMMA
__device__ __forceinline__ v8f wmma4(v2f a, v2f b, v8f c) {
  return __builtin_amdgcn_wmma_f32_16x16x4_f32(false, a, false, b, (short)0, c,
                                               false, false);
}
__device__ __forceinline__ v8f v8zero() {
  v8f z;
  #pragma unroll
  for (int i = 0; i < 8; ++i) z[i] = 0.0f;
  return z;
}
__device__ __forceinline__ float wsum(float v) {
  #pragma unroll
  for (int o = 16; o > 0; o >>= 1) v += __shfl_xor(v, o, 32);
  return v;
}
__device__ __forceinline__ float wmaxr(float v) {
  #pragma unroll
  for (int o = 16; o > 0; o >>= 1) v = fmaxf(v, __shfl_xor(v, o, 32));
  return v;
}

// ---- 1. edge aggregation: c[dst] += h[src]; cnt[dst]++ (one wave per edge)
__global__ void k_edge_agg(const float* __restrict__ h, const int* __restrict__ src,
                           const int* __restrict__ dst, float* __restrict__ c,
                           float* __restrict__ cnt, int E) {
  int wid = threadIdx.x >> 5, lane = threadIdx.x & 31;
  int e = blockIdx.x * 8 + wid;
  if (e >= E) return;
  int s = src[e], d = dst[e];
  const float* hs = h + (size_t)s * DIN;
  float* cd = c + (size_t)d * DIN;
  #pragma unroll
  for (int i = 0; i < 4; ++i) atomicAdd(&cd[lane + 32 * i], hs[lane + 32 * i]);
  if (lane == 0) atomicAdd(&cnt[d], 1.0f);
}

// ---- 2. X = cat(h, c/max(cnt,1))  [N,256]
__global__ void k_build_x(const float* __restrict__ h, const float* __restrict__ c,
                          const float* __restrict__ cnt, float* __restrict__ X) {
  int row = blockIdx.x, col = threadIdx.x;
  float v = (col < DIN) ? h[(size_t)row * DIN + col]
                        : c[(size_t)row * DIN + (col - DIN)] / fmaxf(cnt[row], 1.0f);
  X[(size_t)row * DK + col] = v;
}

// ---- 3. pad W_pool [256,1000]->[256,1024], b_pool [1000]->[1024]
__global__ void k_pad_w(const float* __restrict__ Wp, const float* __restrict__ bp,
                        float* __restrict__ Wpad, float* __restrict__ bpad) {
  int idx = blockIdx.x * 256 + threadIdx.x;
  if (idx < DK * ASSIGN_PAD) {
    int k = idx >> 10, cc = idx & (ASSIGN_PAD - 1);
    Wpad[idx] = (cc < ASSIGNC) ? Wp[k * ASSIGNC + cc] : 0.0f;
  } else if (idx < DK * ASSIGN_PAD + ASSIGN_PAD) {
    int j = idx - DK * ASSIGN_PAD;
    bpad[j] = (j < ASSIGNC) ? bp[j] : 0.0f;
  }
}

// ---- 4. feat = relu(l2norm(X @ W_feat + b))  fused, 16 rows x 128 cols / block
__global__ void __launch_bounds__(256) k_gemm_feat(const float* __restrict__ X,
    const float* __restrict__ W, const float* __restrict__ bias,
    float* __restrict__ feat) {
  __shared__ float zs[16][DIN + 4];
  int wid = threadIdx.x >> 5, lane = threadIdx.x & 31;
  int row0 = blockIdx.x * 16;
  int lm = lane & 15, half = lane >> 4;
  int col0 = wid * 16;
  v8f acc = v8zero();
  const float* arow = X + (size_t)(row0 + lm) * DK + half * 2;
  for (int k0 = 0; k0 < DK; k0 += 4) {
    __builtin_prefetch(arow + k0 + 64, 0, 1);
    v2f a = *(const v2f*)(arow + k0);
    v2f b;
    b.x = W[(size_t)(k0 + half * 2) * DIN + col0 + lm];
    b.y = W[(size_t)(k0 + half * 2 + 1) * DIN + col0 + lm];
    acc = wmma4(a, b, acc);
  }
  float bv = bias[col0 + lm];
  #pragma unroll
  for (int i = 0; i < 8; ++i) zs[i + half * 8][col0 + lm] = acc[i] + bv;
  __syncthreads();
  for (int rr = 0; rr < 2; ++rr) {
    int r = wid * 2 + rr;
    float s = 0.0f;
    #pragma unroll
    for (int i = 0; i < 4; ++i) { float v = zs[r][lane + 32 * i]; s += v * v; }
    s = wsum(s);
    float inv = 1.0f / fmaxf(sqrtf(s), 1e-12f);
    float* fo = feat + (size_t)(row0 + r) * DIN;
    #pragma unroll
    for (int i = 0; i < 4; ++i)
      fo[lane + 32 * i] = fmaxf(zs[r][lane + 32 * i] * inv, 0.0f);
  }
}

// ---- 5. full z = X @ W_pool_pad + b over 1024 cols kept in LDS, then
//         l2norm + relu + exact masked softmax -> compact p_pad [N,64]
__global__ void __launch_bounds__(256) k_gemm_pool(const float* __restrict__ X,
    const float* __restrict__ W, const float* __restrict__ bias,
    float* __restrict__ p) {
  __shared__ float zs[16][ASSIGN_PAD];  // 64KB, fits CDNA5 WGP LDS
  int wid = threadIdx.x >> 5, lane = threadIdx.x & 31;
  int row0 = blockIdx.x * 16;
  int lm = lane & 15, half = lane >> 4;
  int cbase = wid * 128;  // each wave owns 8 column tiles (128 cols)
  v8f acc[8];
  #pragma unroll
  for (int t = 0; t < 8; ++t) acc[t] = v8zero();
  const float* arow = X + (size_t)(row0 + lm) * DK + half * 2;
  for (int k0 = 0; k0 < DK; k0 += 4) {
    __builtin_prefetch(arow + k0 + 64, 0, 1);
    v2f a = *(const v2f*)(arow + k0);
    const float* w0 = W + (size_t)(k0 + half * 2) * ASSIGN_PAD + cbase + lm;
    const float* w1 = w0 + ASSIGN_PAD;
    #pragma unroll
    for (int t = 0; t < 8; ++t) {
      v2f b; b.x = w0[t * 16]; b.y = w1[t * 16];
      acc[t] = wmma4(a, b, acc[t]);
    }
  }
  #pragma unroll
  for (int t = 0; t < 8; ++t) {
    int col = cbase + t * 16 + lm;
    float bv = bias[col];
    #pragma unroll
    for (int i = 0; i < 8; ++i) zs[i + half * 8][col] = acc[t][i] + bv;
  }
  __syncthreads();
  for (int rr = 0; rr < 2; ++rr) {
    int r = wid * 2 + rr;
    int R = row0 + r;
    int g = R / NPER;
    int cb = g * KPER;   // in-block column base
    float ssq = 0.0f;    // norm over full row (padded cols are exact zeros)
    #pragma unroll
    for (int i = 0; i < 32; ++i) { float v = zs[r][lane + 32 * i]; ssq += v * v; }
    ssq = wsum(ssq);
    float inv = 1.0f / fmaxf(sqrtf(ssq), 1e-12f);
    float v0 = fmaxf(zs[r][cb + lane] * inv, 0.0f);          // cols 0..31 (<50)
    int j1 = lane + 32;
    float v1 = (j1 < KPER) ? fmaxf(zs[r][cb + j1] * inv, 0.0f) : 0.0f;
    float m = wmaxr(fmaxf(v0, v1));          // >=0, matches softmax over zeros
    float e0 = expf(v0 - m);
    float e1 = (j1 < KPER) ? expf(v1 - m) : 0.0f;
    float s_in = wsum(e0 + e1);
    float Z = s_in + (float)(ASSIGNC - KPER) * expf(-m);  // 950 masked zeros
    float denom = s_in + 1e-13f * Z;                      // allennlp renorm
    float* pr = p + (size_t)R * PCOLS;
    pr[lane] = e0 / denom;
    pr[j1] = (j1 < KPER) ? e1 / denom : 0.0f;
  }
}

// ---- 6. a_s[dst, :50] += p[src, :50]  (one wave per edge)
__global__ void k_edge_as(const int* __restrict__ src, const int* __restrict__ dst,
                          const float* __restrict__ p, float* __restrict__ a_s,
                          int E) {
  int wid = threadIdx.x >> 5, lane = threadIdx.x & 31;
  int e = blockIdx.x * 8 + wid;
  if (e >= E) return;
  int s = src[e], d = dst[e];
  const float* ps = p + (size_t)s * PCOLS;
  float* ad = a_s + (size_t)d * PCOLS;
  atomicAdd(&ad[lane], ps[lane]);
  int j1 = lane + 32;
  if (j1 < KPER) atomicAdd(&ad[j1], ps[j1]);
}

// ---- 7. per-graph C = P_g^T (64x1000) @ B_g (1000 x ldb); write m<mlim,n<nlim
__global__ void __launch_bounds__(32) k_gemm_atb(const float* __restrict__ P,
    const float* __restrict__ Bm, int ldb, float* __restrict__ out, int ldo,
    int ntn, int mlimit, int nlimit, int outColIsBlock) {
  int g = blockIdx.y;
  int mt = blockIdx.x / ntn, nt = blockIdx.x % ntn;
  int lane = threadIdx.x, lm = lane & 15, half = lane >> 4;
  size_t rowoff = (size_t)g * NPER;
  v8f acc = v8zero();
  for (int k0 = 0; k0 < NPER; k0 += 4) {
    size_t ka = rowoff + k0 + half * 2;
    __builtin_prefetch(P + (ka + 16) * PCOLS, 0, 1);
    v2f a;  // A[m,k] = P[k][m] (transposed access, contiguous over lanes)
    a.x = P[ka * PCOLS + mt * 16 + lm];
    a.y = P[(ka + 1) * PCOLS + mt * 16 + lm];
    v2f b;
    b.x = Bm[ka * (size_t)ldb + nt * 16 + lm];
    b.y = Bm[(ka + 1) * (size_t)ldb + nt * 16 + lm];
    acc = wmma4(a, b, acc);
  }
  int colBase = outColIsBlock ? g * KPER : 0;
  #pragma unroll
  for (int i = 0; i < 8; ++i) {
    int m = mt * 16 + i + half * 8;
    int n = nt * 16 + lm;
    if (m < mlimit && n < nlimit)
      out[(size_t)(g * KPER + m) * ldo + colBase + n] = acc[i];
  }
}

extern "C" void kernel_launch(void* const* d_in, const int* in_sizes, int n_in,
                              void* d_out, int out_size, void* d_ws, size_t ws_size,
                              hipStream_t stream) {
  const float* h      = (const float*)d_in[0];
  const float* W_feat = (const float*)d_in[1];
  const float* b_feat = (const float*)d_in[2];
  const float* W_pool = (const float*)d_in[3];
  const float* b_pool = (const float*)d_in[4];
  const int*   src    = (const int*)d_in[5];
  const int*   dst    = (const int*)d_in[6];
  int E = in_sizes[5];

  float* ws   = (float*)d_ws;
  float* c    = ws;                                    // N*128
  float* cnt  = c + (size_t)N_NODES * DIN;             // N
  float* X    = cnt + N_NODES;                         // N*256
  float* Wp   = X + (size_t)N_NODES * DK;              // 256*1024
  float* bp   = Wp + (size_t)DK * ASSIGN_PAD;          // 1024
  float* feat = bp + ASSIGN_PAD;                       // N*128
  float* p    = feat + (size_t)N_NODES * DIN;          // N*64
  float* a_s  = p + (size_t)N_NODES * PCOLS;           // N*64

  float* outAdj = (float*)d_out;                       // [1000,1000]
  float* outHp  = outAdj + (size_t)ASSIGNC * ASSIGNC;  // [1000,128]

  (void)hipMemsetAsync(c,   0, (size_t)N_NODES * DIN * sizeof(float), stream);
  (void)hipMemsetAsync(cnt, 0, (size_t)N_NODES * sizeof(float), stream);
  (void)hipMemsetAsync(a_s, 0, (size_t)N_NODES * PCOLS * sizeof(float), stream);
  (void)hipMemsetAsync(outAdj, 0, (size_t)ASSIGNC * ASSIGNC * sizeof(float), stream);

  k_edge_agg<<<(E + 7) / 8, 256, 0, stream>>>(h, src, dst, c, cnt, E);
  k_build_x<<<N_NODES, 256, 0, stream>>>(h, c, cnt, X);
  k_pad_w<<<(DK * ASSIGN_PAD + ASSIGN_PAD + 255) / 256, 256, 0, stream>>>(
      W_pool, b_pool, Wp, bp);
  k_gemm_feat<<<N_NODES / 16, 256, 0, stream>>>(X, W_feat, b_feat, feat);
  k_gemm_pool<<<N_NODES / 16, 256, 0, stream>>>(X, Wp, bp, p);
  k_edge_as<<<(E + 7) / 8, 256, 0, stream>>>(src, dst, p, a_s, E);
  // h_pool = p^T @ feat  -> [1000,128] at outHp
  k_gemm_atb<<<dim3(4 * 8, NGRAPH), 32, 0, stream>>>(p, feat, DIN, outHp, DIN,
                                                     8, KPER, DIN, 0);
  // adj = p^T @ a_s -> block-diagonal [1000,1000] at outAdj
  k_gemm_atb<<<dim3(4 * 4, NGRAPH), 32, 0, stream>>>(p, a_s, PCOLS, outAdj,
                                                     ASSIGNC, 4, KPER, KPER, 1);
}